// NGCNLayer_42614665511369
// MI455X (gfx1250) — compile-verified
//
#include <hip/hip_runtime.h>

typedef __attribute__((ext_vector_type(2))) float v2f;
typedef __attribute__((ext_vector_type(8))) float v8f;

#define KDIM 512
#define NDIM 128
#define APITCH 516   // 516 % 64 = 4 -> 16 lanes hit distinct bank pairs on b64 reads

// ---------------------------------------------------------------------------
// Dense GEMM: out[M,128] = A[M,512] @ W[512,128] using V_WMMA_F32_16X16X4_F32.
// Block = 256 threads = 8 waves. Block owns one 16-row M tile (staged in LDS),
// wave w owns N-tile w (16 cols). 128 WMMA steps of K=4 each.
// ---------------------------------------------------------------------------
__global__ __launch_bounds__(256)
void gemm_wmma_f32(const float* __restrict__ A, const float* __restrict__ W,
                   float* __restrict__ out) {
    __shared__ float As[16 * APITCH];

    const int mbase = blockIdx.x * 16;
    const float* Ablk = A + (size_t)mbase * KDIM;   // 16 rows are contiguous (32KB)

    // Cooperative, coalesced float4 copy of the A tile into padded LDS.
    for (int i = threadIdx.x; i < (16 * KDIM) / 4; i += 256) {
        const int idx = i * 4;
        const int r = idx >> 9;        // / 512
        const int c = idx & (KDIM - 1);
        const float4 v = *(const float4*)(Ablk + idx);
        float* dst = &As[r * APITCH + c];
        dst[0] = v.x; dst[1] = v.y; dst[2] = v.z; dst[3] = v.w;
    }
    __syncthreads();

    const int wave = threadIdx.x >> 5;     // 0..7 -> N tile
    const int lane = threadIdx.x & 31;
    const int lm   = lane & 15;            // M (for A) / N (for B) index
    const int lh   = lane >> 4;            // K half-select
    const int nbase = wave * 16;

    // A frag: a.x = A[m=lm][k + 2*lh], a.y = A[m=lm][k + 2*lh + 1]
    const float* arow = &As[lm * APITCH + 2 * lh];
    // B frag: b.x = W[k + 2*lh][nbase+lm], b.y = W[k + 2*lh + 1][nbase+lm]
    const float* bcol = W + (size_t)(2 * lh) * NDIM + nbase + lm;

    v8f acc = {0.f, 0.f, 0.f, 0.f, 0.f, 0.f, 0.f, 0.f};

#pragma unroll 4
    for (int k = 0; k < KDIM; k += 4) {
        v2f a, b;
        a.x = arow[k];
        a.y = arow[k + 1];
        b.x = bcol[(size_t)k * NDIM];
        b.y = bcol[(size_t)(k + 1) * NDIM];
        // (neg_a, A, neg_b, B, c_mod, C, reuse_a, reuse_b)
        acc = __builtin_amdgcn_wmma_f32_16x16x4_f32(false, a, false, b,
                                                    (short)0, acc, false, false);
    }

    // D layout: VGPR r -> row (r + 8*lh), col = lm
    float* o = out + (size_t)(mbase + lh * 8) * NDIM + nbase + lm;
#pragma unroll
    for (int r = 0; r < 8; ++r)
        o[(size_t)r * NDIM] = acc[r];
}

// ---------------------------------------------------------------------------
// SpMM: out[row[e]] += val[e] * x[col[e]].  One wave per edge; lane handles
// 4 features -> coalesced 512B row gather + native f32 atomics (L2-resident).
// ---------------------------------------------------------------------------
__global__ __launch_bounds__(256)
void spmm_edges(const float* __restrict__ x, const float* __restrict__ val,
                const int* __restrict__ row, const int* __restrict__ col,
                float* __restrict__ out, int E) {
    const int lane = threadIdx.x & 31;
    const int wid  = (blockIdx.x * blockDim.x + threadIdx.x) >> 5;
    const int nw   = (gridDim.x * blockDim.x) >> 5;

    for (int e = wid; e < E; e += nw) {
        const int en = e + nw;
        if (en < E) {  // warm L2 line for the next gathered row
            __builtin_prefetch(x + (size_t)col[en] * NDIM + lane * 4, 0, 0);
        }
        const int   r = row[e];
        const int   c = col[e];
        const float v = val[e];
        const float4 xv = *(const float4*)(x + (size_t)c * NDIM + lane * 4);
        float* o = out + (size_t)r * NDIM + lane * 4;
        unsafeAtomicAdd(o + 0, xv.x * v);
        unsafeAtomicAdd(o + 1, xv.y * v);
        unsafeAtomicAdd(o + 2, xv.z * v);
        unsafeAtomicAdd(o + 3, xv.w * v);
    }
}

// ---------------------------------------------------------------------------
__global__ void zero_f4(float4* __restrict__ p, long n4) {
    long i = blockIdx.x * (long)blockDim.x + threadIdx.x;
    const long stride = (long)gridDim.x * blockDim.x;
    const float4 z = make_float4(0.f, 0.f, 0.f, 0.f);
    for (; i < n4; i += stride) p[i] = z;
}

__global__ void bias_add_f4(const float4* __restrict__ s, const float4* __restrict__ b,
                            float4* __restrict__ o, long n4) {
    long i = blockIdx.x * (long)blockDim.x + threadIdx.x;
    const long stride = (long)gridDim.x * blockDim.x;
    for (; i < n4; i += stride) {
        const float4 sv = s[i];
        const float4 bv = b[i & 31];            // 128 floats = 32 float4 per row
        o[i] = make_float4(sv.x + bv.x, sv.y + bv.y, sv.z + bv.z, sv.w + bv.w);
    }
}

// ---------------------------------------------------------------------------
extern "C" void kernel_launch(void* const* d_in, const int* in_sizes, int n_in,
                              void* d_out, int out_size, void* d_ws, size_t ws_size,
                              hipStream_t stream) {
    const float* input  = (const float*)d_in[0];   // [N, 512]
    const float* weight = (const float*)d_in[1];   // [512, 128]
    const float* bias   = (const float*)d_in[2];   // [128]
    const float* eval   = (const float*)d_in[3];   // [E]
    const int*   erow   = (const int*)d_in[4];     // [E]
    const int*   ecol   = (const int*)d_in[5];     // [E]

    const int E = in_sizes[3];
    const int M = in_sizes[0] / KDIM;              // 100000 (multiple of 16)
    float* out = (float*)d_out;
    float* ws  = (float*)d_ws;                     // one [M,128] ping buffer

    const long n  = (long)M * NDIM;
    const long n4 = n / 4;

    dim3 blk(256);

    // support0 = input @ weight  -> out
    gemm_wmma_f32<<<M / 16, blk, 0, stream>>>(input, weight, out);

    // iter 1: ws = spmm(out)
    zero_f4<<<1024, blk, 0, stream>>>((float4*)ws, n4);
    spmm_edges<<<4096, blk, 0, stream>>>(out, eval, erow, ecol, ws, E);

    // iter 2: out = spmm(ws)
    zero_f4<<<1024, blk, 0, stream>>>((float4*)out, n4);
    spmm_edges<<<4096, blk, 0, stream>>>(ws, eval, erow, ecol, out, E);

    // iter 3: ws = spmm(out)
    zero_f4<<<1024, blk, 0, stream>>>((float4*)ws, n4);
    spmm_edges<<<4096, blk, 0, stream>>>(out, eval, erow, ecol, ws, E);

    // out = ws + bias
    bias_add_f4<<<1024, blk, 0, stream>>>((const float4*)ws, (const float4*)bias,
                                          (float4*)out, n4);
}